// pointnet2_feat_msg_73667279061537
// MI455X (gfx1250) — compile-verified
//
#include <hip/hip_runtime.h>
#include <hip/hip_bf16.h>
#include <stdint.h>

typedef float v2f __attribute__((ext_vector_type(2)));
typedef float v4f __attribute__((ext_vector_type(4)));
typedef float v8f __attribute__((ext_vector_type(8)));

#define WAVE 32

// ---------------------------------------------------------------------------
// Farthest-point sampling: one workgroup per batch. All coords + running
// min-dist live in LDS (4*N floats; SA1: 256KB < 320KB WGP LDS).
// ---------------------------------------------------------------------------
__global__ void fps_lds_kernel(const float* __restrict__ pts, int N, int M,
                               int pstride, size_t batchStride,
                               int* __restrict__ outIdx,
                               float* __restrict__ outXyz) {
  extern __shared__ float smem[];
  float* px   = smem;
  float* py   = px + N;
  float* pz   = py + N;
  float* dist = pz + N;
  float* wmax = dist + N;            // [32]
  int*   widx = (int*)(wmax + 32);   // [32]
  int*   scur = widx + 32;           // [1]

  const int b = blockIdx.x;
  const int tid = threadIdx.x;
  const int nthr = blockDim.x;
  const int nw = nthr / WAVE;
  const float* base = pts + (size_t)b * batchStride;

  for (int p = tid; p < N; p += nthr) {
    px[p] = base[(size_t)p * pstride + 0];
    py[p] = base[(size_t)p * pstride + 1];
    pz[p] = base[(size_t)p * pstride + 2];
    dist[p] = 1e10f;
  }
  __syncthreads();

  int cur = 0;
  for (int it = 0; it < M; ++it) {
    if (tid == 0) {
      outIdx[b * M + it] = cur;
      outXyz[((size_t)b * M + it) * 3 + 0] = px[cur];
      outXyz[((size_t)b * M + it) * 3 + 1] = py[cur];
      outXyz[((size_t)b * M + it) * 3 + 2] = pz[cur];
    }
    const float cx = px[cur], cy = py[cur], cz = pz[cur];
    float best = -1.0f; int bi = N;
    for (int p = tid; p < N; p += nthr) {
      const float dx = px[p] - cx, dy = py[p] - cy, dz = pz[p] - cz;
      const float d = dx * dx + dy * dy + dz * dz;
      const float nd = fminf(dist[p], d);
      dist[p] = nd;
      if (nd > best) { best = nd; bi = p; }
    }
    for (int off = WAVE / 2; off > 0; off >>= 1) {
      const float ov = __shfl_xor(best, off, WAVE);
      const int   oi = __shfl_xor(bi,   off, WAVE);
      if (ov > best || (ov == best && oi < bi)) { best = ov; bi = oi; }
    }
    if ((tid & (WAVE - 1)) == 0) { wmax[tid >> 5] = best; widx[tid >> 5] = bi; }
    __syncthreads();
    if (tid == 0) {
      float bb = wmax[0]; int bbi = widx[0];
      for (int w = 1; w < nw; ++w)
        if (wmax[w] > bb || (wmax[w] == bb && widx[w] < bbi)) { bb = wmax[w]; bbi = widx[w]; }
      scur[0] = bbi;
    }
    __syncthreads();
    cur = scur[0];
  }
}

// ---------------------------------------------------------------------------
// Ball query, one wave32 per (batch, center): first K points (ascending
// index) within radius; pad with first hit; all 3 radii in one scan.
// ---------------------------------------------------------------------------
__global__ void ballquery_kernel(const float* __restrict__ pts, int N, int pstride,
                                 size_t batchStride,
                                 const float* __restrict__ centers, int M, int Btot,
                                 float r0sq, float r1sq, float r2sq,
                                 int K0, int K1, int K2,
                                 int* __restrict__ idx0, int* __restrict__ idx1,
                                 int* __restrict__ idx2) {
  const int gw = (int)((blockIdx.x * blockDim.x + threadIdx.x) >> 5);
  if (gw >= Btot * M) return;
  const int lane = threadIdx.x & (WAVE - 1);
  const int b = gw / M;
  const float* base = pts + (size_t)b * batchStride;
  const float qx = centers[(size_t)gw * 3 + 0];
  const float qy = centers[(size_t)gw * 3 + 1];
  const float qz = centers[(size_t)gw * 3 + 2];

  int c0 = 0, c1 = 0, c2 = 0;
  int f0 = 0, f1 = 0, f2 = 0;
  for (int bs = 0; bs < N; bs += WAVE) {
    const int p = bs + lane;
    float d2 = 1e30f;
    if (p < N) {
      const float dx = base[(size_t)p * pstride + 0] - qx;
      const float dy = base[(size_t)p * pstride + 1] - qy;
      const float dz = base[(size_t)p * pstride + 2] - qz;
      d2 = dx * dx + dy * dy + dz * dz;
    }
    unsigned m0 = (unsigned)__ballot(d2 <= r0sq);
    unsigned m1 = (unsigned)__ballot(d2 <= r1sq);
    unsigned m2 = (unsigned)__ballot(d2 <= r2sq);
    while (m0 && c0 < K0) {
      const int pi = bs + __builtin_ctz(m0);
      if (c0 == 0) f0 = pi;
      if (lane == 0) idx0[(size_t)gw * K0 + c0] = pi;
      ++c0; m0 &= m0 - 1;
    }
    while (m1 && c1 < K1) {
      const int pi = bs + __builtin_ctz(m1);
      if (c1 == 0) f1 = pi;
      if (lane == 0) idx1[(size_t)gw * K1 + c1] = pi;
      ++c1; m1 &= m1 - 1;
    }
    while (m2 && c2 < K2) {
      const int pi = bs + __builtin_ctz(m2);
      if (c2 == 0) f2 = pi;
      if (lane == 0) idx2[(size_t)gw * K2 + c2] = pi;
      ++c2; m2 &= m2 - 1;
    }
    if (c0 >= K0 && c1 >= K1 && c2 >= K2) break;
  }
  if (lane == 0) {
    for (int i = c0; i < K0; ++i) idx0[(size_t)gw * K0 + i] = f0;
    for (int i = c1; i < K1; ++i) idx1[(size_t)gw * K1 + i] = f1;
    for (int i = c2; i < K2; ++i) idx2[(size_t)gw * K2 + i] = f2;
  }
}

// ---------------------------------------------------------------------------
// Weight prep: Wp[o][c] = W[o][c] * s[o], zero-padded to CiPad columns.
// Removes all guards and the scale multiply from the GEMM hot loop.
// ---------------------------------------------------------------------------
__global__ void prep_w_kernel(const float* __restrict__ W, const float* __restrict__ s,
                              float* __restrict__ Wp, int Co, int Ci, int CiPad,
                              int total) {
  const int t = blockIdx.x * blockDim.x + threadIdx.x;
  if (t >= total) return;
  const int c = t % CiPad;
  const int o = t / CiPad;
  Wp[t] = (c < Ci) ? W[(size_t)o * Ci + c] * s[o] : 0.0f;
}

// ---------------------------------------------------------------------------
// Gather SA1: row-major (R x 8) per batch (6 real channels + 2 zero pad).
// ---------------------------------------------------------------------------
__global__ void gather_sa1_kernel(const float* __restrict__ x, int N,
                                  const float* __restrict__ centers,
                                  const int* __restrict__ idx, int M, int K,
                                  float* __restrict__ Fout, int total) {
  const int t = blockIdx.x * blockDim.x + threadIdx.x;
  if (t >= total) return;  // total = B * R * 8
  const int c = t & 7;
  const int r = (t >> 3) % (M * K);
  const int b = t / (8 * M * K);
  float v = 0.0f;
  if (c < 6) {
    const int pi = idx[(size_t)b * M * K + r];
    v = x[((size_t)b * N + pi) * 6 + c];
    if (c < 3) v -= centers[((size_t)b * M + r / K) * 3 + c];
  }
  Fout[t] = v;
}

// ---------------------------------------------------------------------------
// Gather SA2: row-major (R x 324) per batch (3 rel-xyz + 320 feats + 1 pad).
// ---------------------------------------------------------------------------
__global__ void gather_sa2_kernel(const float* __restrict__ xyz1, int Npts,
                                  const float* __restrict__ feats, int Cf,
                                  const float* __restrict__ centers,
                                  const int* __restrict__ idx, int M, int K,
                                  float* __restrict__ Fout, int Ci, int CiPad,
                                  int total) {
  const int t = blockIdx.x * blockDim.x + threadIdx.x;
  if (t >= total) return;  // total = B * R * CiPad
  const int c = t % CiPad;
  const int r = (t / CiPad) % (M * K);
  const int b = t / (CiPad * M * K);
  float v = 0.0f;
  if (c < Ci) {
    const int pi = idx[(size_t)b * M * K + r];
    if (c < 3) v = xyz1[((size_t)b * Npts + pi) * 3 + c] -
                   centers[((size_t)b * M + r / K) * 3 + c];
    else       v = feats[((size_t)b * Npts + pi) * Cf + (c - 3)];
  }
  Fout[t] = v;
}

// SA3 input: row-major (128 x 644) per batch: concat(new_xyz2, pts2) + 1 pad.
__global__ void build_sa3_kernel(const float* __restrict__ xyz2,
                                 const float* __restrict__ pts2, int Cf,
                                 float* __restrict__ Fout, int Ci, int CiPad,
                                 int P, int total) {
  const int t = blockIdx.x * blockDim.x + threadIdx.x;
  if (t >= total) return;  // total = B * P * CiPad
  const int c = t % CiPad;
  const int p = (t / CiPad) % P;
  const int b = t / (CiPad * P);
  float v = 0.0f;
  if (c < 3)       v = xyz2[((size_t)b * P + p) * 3 + c];
  else if (c < Ci) v = pts2[((size_t)b * P + p) * Cf + (c - 3)];
  Fout[t] = v;
}

// ---------------------------------------------------------------------------
// WMMA f32 GEMM. One wave per 16x16 tile of (R rows x Co cols), K-loop of 4
// via V_WMMA_F32_16X16X4_F32. Straight-line inner loop: 2x b64 loads + WMMA.
// A (16x4): lanes 0-15 M=lane, regs K0/K1; lanes 16-31 regs K2/K3 (mirrored B).
// C/D: reg j -> M=j+8*half, N=lane%16 => 8 contiguous outputs per lane ->
// two b128 stores, row-major output feeding the next layer directly.
// Co, R multiples of 16; CiPad multiple of 4; pads are zero.
// ---------------------------------------------------------------------------
__global__ void gemm_wmma_f32_kernel(const float* __restrict__ Wp,
                                     const float* __restrict__ bias,
                                     const float* __restrict__ Fin,
                                     float* __restrict__ Fout,
                                     int Co, int CiPad, int R, int relu) {
  const int tilesO = Co >> 4;
  const int tilesR = R >> 4;
  const int t = blockIdx.x * (blockDim.x >> 5) + (threadIdx.x >> 5);
  if (t >= tilesO * tilesR) return;  // wave-uniform; EXEC stays all-ones
  const int b = blockIdx.y;
  const float* fin = Fin + (size_t)b * CiPad * R;
  float* fout = Fout + (size_t)b * Co * R;

  const int to = t / tilesR, tr = t % tilesR;
  const int o0 = to << 4, r0 = tr << 4;
  const int lane = threadIdx.x & (WAVE - 1);
  const int half = lane >> 4;
  const int lm = lane & 15;

  const float* wrow = Wp  + (size_t)(o0 + lm) * CiPad + 2 * half;
  const float* frow = fin + (size_t)(r0 + lm) * CiPad + 2 * half;

  v8f acc = {0.f, 0.f, 0.f, 0.f, 0.f, 0.f, 0.f, 0.f};
  for (int c0 = 0; c0 < CiPad; c0 += 4) {
    const v2f A  = *(const v2f*)(wrow + c0);
    const v2f Bv = *(const v2f*)(frow + c0);
    acc = __builtin_amdgcn_wmma_f32_16x16x4_f32(false, A, false, Bv,
                                                (short)0, acc, false, false);
  }

  const int obase = o0 + 8 * half;
  float o[8];
#pragma unroll
  for (int j = 0; j < 8; ++j) {
    float v = acc[j];
    if (bias) v += bias[obase + j];
    if (relu) v = fmaxf(v, 0.0f);
    o[j] = v;
  }
  float* dst = fout + (size_t)(r0 + lm) * Co + obase;
  v4f lo = {o[0], o[1], o[2], o[3]};
  v4f hi = {o[4], o[5], o[6], o[7]};
  *(v4f*)(dst)     = lo;
  *(v4f*)(dst + 4) = hi;
}

// Max-pool over K neighbors: row-major (M*K x Co) -> point-major out slab.
__global__ void poolk_kernel(const float* __restrict__ in, int Co, int M, int K,
                             float* __restrict__ out, int coff, int ctot, int total) {
  const int t = blockIdx.x * blockDim.x + threadIdx.x;
  if (t >= total) return;  // total = B * M * Co
  const int o = t % Co;
  const int m = (t / Co) % M;
  const int b = t / (Co * M);
  const float* p = in + ((size_t)b * M * K + (size_t)m * K) * Co + o;
  float v = p[0];
  for (int k = 1; k < K; ++k) v = fmaxf(v, p[(size_t)k * Co]);
  out[((size_t)b * M + m) * ctot + coff + o] = v;
}

// SA3 max-pool over points: row-major (P x C) per batch -> (B x C) row-major.
__global__ void poolp_kernel(const float* __restrict__ in, int C, int P,
                             float* __restrict__ out, int total) {
  const int t = blockIdx.x * blockDim.x + threadIdx.x;
  if (t >= total) return;  // total = B * C
  const int c = t % C;
  const int b = t / C;
  const float* p = in + (size_t)b * P * C + c;
  float v = p[0];
  for (int i = 1; i < P; ++i) v = fmaxf(v, p[(size_t)i * C]);
  out[t] = v;
}

// ---------------------------------------------------------------------------
static inline void launch_gemm(const float* Wp, const float* bi,
                               const float* fin, float* fout, int Co, int CiPad,
                               int R, int relu, int batches, hipStream_t stream) {
  const int tiles = (Co / 16) * (R / 16);
  dim3 grid((tiles + 3) / 4, batches);
  gemm_wmma_f32_kernel<<<grid, 128, 0, stream>>>(Wp, bi, fin, fout, Co, CiPad, R, relu);
}

extern "C" void kernel_launch(void* const* d_in, const int* in_sizes, int n_in,
                              void* d_out, int out_size, void* d_ws, size_t ws_size,
                              hipStream_t stream) {
  (void)in_sizes; (void)n_in; (void)out_size; (void)ws_size;
  const int B = 16, N = 16384;
  const float* x = (const float*)d_in[0];
  const float* fc1w = (const float*)d_in[64];
  auto inp = [&](int i) { return (const float*)d_in[i]; };

  // ---- workspace carve (deterministic order every call) ----
  char* ws = (char*)d_ws;
  size_t off = 0;
  auto carve = [&](size_t bytes) -> void* {
    void* p = ws + off;
    off += (bytes + 255) & ~(size_t)255;
    return p;
  };
  const size_t SLAB = (size_t)128 * 4096;  // max per-batch (rows*CiPad) floats
  float* bufs[2];
  bufs[0] = (float*)carve(sizeof(float) * (size_t)B * SLAB);
  bufs[1] = (float*)carve(sizeof(float) * (size_t)B * SLAB);
  float* newxyz1 = (float*)carve(sizeof(float) * (size_t)B * 512 * 3);
  float* newxyz2 = (float*)carve(sizeof(float) * (size_t)B * 128 * 3);
  int* fpsidx1 = (int*)carve(sizeof(int) * (size_t)B * 512);
  int* fpsidx2 = (int*)carve(sizeof(int) * (size_t)B * 128);
  int* bq1[3]; int* bq2[3];
  const int Ks[3] = {2, 4, 8};
  for (int j = 0; j < 3; ++j) bq1[j] = (int*)carve(sizeof(int) * (size_t)B * 512 * Ks[j]);
  for (int j = 0; j < 3; ++j) bq2[j] = (int*)carve(sizeof(int) * (size_t)B * 128 * Ks[j]);
  float* pts1 = (float*)carve(sizeof(float) * (size_t)B * 512 * 320);
  float* pts2 = (float*)carve(sizeof(float) * (size_t)B * 128 * 640);
  float* featfc = (float*)carve(sizeof(float) * (size_t)B * 1024);

  // padded+scale-folded weights for the 21 MLP layers
  const int sa1_w[3][3] = {{32, 32, 64}, {64, 64, 128}, {64, 96, 128}};
  const int sa2_w[3][3] = {{64, 64, 128}, {128, 128, 256}, {128, 128, 256}};
  const int sa3_w[3] = {256, 512, 1024};
  float* wp1[3][3]; float* wp2[3][3]; float* wp3[3];
  int cip1[3][3], cip2[3][3], cip3[3];
  for (int j = 0; j < 3; ++j)
    for (int l = 0; l < 3; ++l) {
      cip1[j][l] = (l == 0) ? 8 : sa1_w[j][l - 1];
      wp1[j][l] = (float*)carve(sizeof(float) * (size_t)sa1_w[j][l] * cip1[j][l]);
    }
  for (int j = 0; j < 3; ++j)
    for (int l = 0; l < 3; ++l) {
      cip2[j][l] = (l == 0) ? 324 : sa2_w[j][l - 1];
      wp2[j][l] = (float*)carve(sizeof(float) * (size_t)sa2_w[j][l] * cip2[j][l]);
    }
  for (int l = 0; l < 3; ++l) {
    cip3[l] = (l == 0) ? 644 : sa3_w[l - 1];
    wp3[l] = (float*)carve(sizeof(float) * (size_t)sa3_w[l] * cip3[l]);
  }

  // ---- weight prep (scale folded, zero pad) ----
  auto prep = [&](const float* W, const float* s, float* Wp, int Co, int Ci, int CiPad) {
    const int total = Co * CiPad;
    prep_w_kernel<<<(total + 255) / 256, 256, 0, stream>>>(W, s, Wp, Co, Ci, CiPad, total);
  };
  for (int j = 0; j < 3; ++j)
    for (int l = 0; l < 3; ++l) {
      const int base = 1 + j * 9 + l * 3;
      prep(inp(base), inp(base + 1), wp1[j][l], sa1_w[j][l],
           (l == 0) ? 6 : sa1_w[j][l - 1], cip1[j][l]);
    }
  for (int j = 0; j < 3; ++j)
    for (int l = 0; l < 3; ++l) {
      const int base = 28 + j * 9 + l * 3;
      prep(inp(base), inp(base + 1), wp2[j][l], sa2_w[j][l],
           (l == 0) ? 323 : sa2_w[j][l - 1], cip2[j][l]);
    }
  for (int l = 0; l < 3; ++l) {
    const int base = 55 + l * 3;
    prep(inp(base), inp(base + 1), wp3[l], sa3_w[l],
         (l == 0) ? 643 : sa3_w[l - 1], cip3[l]);
  }

  // ---- SA1 ----
  {
    size_t shmem = (4 * (size_t)N + 32) * sizeof(float) + 33 * sizeof(int);
    fps_lds_kernel<<<B, 256, shmem, stream>>>(x, N, 512, 6, (size_t)N * 6,
                                              fpsidx1, newxyz1);
  }
  {
    const int waves = B * 512;
    ballquery_kernel<<<(waves * 32 + 127) / 128, 128, 0, stream>>>(
        x, N, 6, (size_t)N * 6, newxyz1, 512, B,
        0.01f, 0.04f, 0.16f, 2, 4, 8, bq1[0], bq1[1], bq1[2]);
  }
  int coff = 0;
  for (int j = 0; j < 3; ++j) {
    const int K = Ks[j], M = 512, R = M * K;
    const int gtotal = B * R * 8;
    gather_sa1_kernel<<<(gtotal + 255) / 256, 256, 0, stream>>>(
        x, N, newxyz1, bq1[j], M, K, bufs[0], gtotal);
    int cursrc = 0;
    for (int l = 0; l < 3; ++l) {
      launch_gemm(wp1[j][l], inp(1 + j * 9 + l * 3 + 2),
                  bufs[cursrc], bufs[1 - cursrc], sa1_w[j][l], cip1[j][l], R, 1, B, stream);
      cursrc ^= 1;
    }
    const int co = sa1_w[j][2];
    const int ptotal = B * M * co;
    poolk_kernel<<<(ptotal + 255) / 256, 256, 0, stream>>>(
        bufs[cursrc], co, M, K, pts1, coff, 320, ptotal);
    coff += co;
  }

  // ---- SA2 ----
  {
    size_t shmem = (4 * (size_t)512 + 32) * sizeof(float) + 33 * sizeof(int);
    fps_lds_kernel<<<B, 256, shmem, stream>>>(newxyz1, 512, 128, 3, (size_t)512 * 3,
                                              fpsidx2, newxyz2);
  }
  {
    const int waves = B * 128;
    ballquery_kernel<<<(waves * 32 + 127) / 128, 128, 0, stream>>>(
        newxyz1, 512, 3, (size_t)512 * 3, newxyz2, 128, B,
        0.04f, 0.16f, 0.64f, 2, 4, 8, bq2[0], bq2[1], bq2[2]);
  }
  coff = 0;
  for (int j = 0; j < 3; ++j) {
    const int K = Ks[j], M = 128, R = M * K;
    const int gtotal = B * R * 324;
    gather_sa2_kernel<<<(gtotal + 255) / 256, 256, 0, stream>>>(
        newxyz1, 512, pts1, 320, newxyz2, bq2[j], M, K, bufs[0], 323, 324, gtotal);
    int cursrc = 0;
    for (int l = 0; l < 3; ++l) {
      launch_gemm(wp2[j][l], inp(28 + j * 9 + l * 3 + 2),
                  bufs[cursrc], bufs[1 - cursrc], sa2_w[j][l], cip2[j][l], R, 1, B, stream);
      cursrc ^= 1;
    }
    const int co = sa2_w[j][2];
    const int ptotal = B * M * co;
    poolk_kernel<<<(ptotal + 255) / 256, 256, 0, stream>>>(
        bufs[cursrc], co, M, K, pts2, coff, 640, ptotal);
    coff += co;
  }

  // ---- SA3: 643 -> 256 -> 512 -> 1024 over 128 points, then max-pool ----
  {
    const int P = 128;
    const int gtotal = B * P * 644;
    build_sa3_kernel<<<(gtotal + 255) / 256, 256, 0, stream>>>(
        newxyz2, pts2, 640, bufs[0], 643, 644, P, gtotal);
    int cursrc = 0;
    for (int l = 0; l < 3; ++l) {
      launch_gemm(wp3[l], inp(55 + l * 3 + 2),
                  bufs[cursrc], bufs[1 - cursrc], sa3_w[l], cip3[l], P, 1, B, stream);
      cursrc ^= 1;
    }
    const int ptotal = B * 1024;
    poolp_kernel<<<(ptotal + 255) / 256, 256, 0, stream>>>(
        bufs[cursrc], 1024, P, featfc, ptotal);
  }

  // ---- Final FC: (B x 1024) @ fc1_w^T -> d_out (B x 512), direct write ----
  launch_gemm(fc1w, nullptr, featfc, (float*)d_out, 512, 1024, 16, 0, 1, stream);
}